// AwesomeRnn_9216999817260
// MI455X (gfx1250) — compile-verified
//
#include <hip/hip_runtime.h>

#define SEQ   256
#define BATCH 128
#define DIMK  1024
#define UNITS 1024
#define GATES 4096
#define SB    (SEQ * BATCH)

#define NBLK_REC    256
#define WHH_STRIDE  1028   // 1024 + 4 pad -> conflict-free b64 LDS reads across 64 banks
#define GBUF_STRIDE 68
#define LDS_FLOATS  (64 * WHH_STRIDE + 32 * GBUF_STRIDE + 32 * 16 + 32)

typedef float v2f __attribute__((ext_vector_type(2)));
typedef float v8f __attribute__((ext_vector_type(8)));

__device__ __forceinline__ v8f wmma4(v2f a, v2f b, v8f c) {
  // D = A(16x4) * B(4x16) + C, fp32 WMMA
  return __builtin_amdgcn_wmma_f32_16x16x4_f32(false, a, false, b, (short)0, c, false, false);
}

__device__ __forceinline__ float sigm(float v) { return 1.0f / (1.0f + __expf(-v)); }
__device__ __forceinline__ float tanh_fast(float v) {
  v = fminf(fmaxf(v, -15.0f), 15.0f);
  float e = __expf(2.0f * v);
  return (e - 1.0f) / (e + 1.0f);
}

// ---- init: zero the per-step barrier counters, seed h ping buffer 0 with h0
__global__ void __launch_bounds__(256) init_kernel(const float* __restrict__ h0,
                                                   float* __restrict__ hping,
                                                   unsigned* __restrict__ counters) {
  int i = blockIdx.x * blockDim.x + threadIdx.x;
  if (i < BATCH * UNITS) hping[i] = h0[i];
  if (i < SEQ) counters[i] = 0u;
}

// ---- Phase 1: xg = x @ W_ih^T   (SB x DIMK) * (DIMK x GATES), fp32 WMMA.
// block = 256 threads (8 waves) -> 128x64 output tile; each wave owns a 2x2 tile
// quad (32x32), so 4 fragment loads feed 4 WMMAs (1.0 b64 load per WMMA).
__global__ void __launch_bounds__(256) xg_gemm_kernel(const float* __restrict__ x,
                                                      const float* __restrict__ Wih,
                                                      float* __restrict__ xg) {
  const int lane = threadIdx.x & 31;
  const int wave = threadIdx.x >> 5;     // 0..7
  const int rp   = wave & 3;             // row-tile pair: tiles 2rp, 2rp+1
  const int cp   = wave >> 2;            // col-tile pair: tiles 2cp, 2cp+1
  const int rowBase = blockIdx.x * 128 + rp * 32;
  const int colBase = blockIdx.y * 64 + cp * 32;
  const int arow = lane & 15;
  const int khi  = (lane >> 4) << 1;     // lanes 16-31 hold K+2..K+3

  const float* ap0 = x   + (size_t)(rowBase + arow) * DIMK + khi;
  const float* ap1 = x   + (size_t)(rowBase + 16 + arow) * DIMK + khi;
  const float* bp0 = Wih + (size_t)(colBase + arow) * DIMK + khi;       // W row = output col
  const float* bp1 = Wih + (size_t)(colBase + 16 + arow) * DIMK + khi;

  v8f acc00 = {}; v8f acc01 = {}; v8f acc10 = {}; v8f acc11 = {};
#pragma unroll 4
  for (int kb = 0; kb < DIMK; kb += 4) {
    v2f a0 = *(const v2f*)(ap0 + kb);
    v2f a1 = *(const v2f*)(ap1 + kb);
    v2f b0 = *(const v2f*)(bp0 + kb);
    v2f b1 = *(const v2f*)(bp1 + kb);
    acc00 = wmma4(a0, b0, acc00);
    acc01 = wmma4(a0, b1, acc01);
    acc10 = wmma4(a1, b0, acc10);
    acc11 = wmma4(a1, b1, acc11);
  }
  const int mo = (lane >> 4) << 3;       // C/D: lanes 16-31 hold M+8
#pragma unroll
  for (int v = 0; v < 8; ++v) {
    size_t r0 = (size_t)(rowBase + v + mo);
    size_t r1 = (size_t)(rowBase + 16 + v + mo);
    xg[r0 * GATES + colBase + arow]      = acc00[v];
    xg[r0 * GATES + colBase + 16 + arow] = acc01[v];
    xg[r1 * GATES + colBase + arow]      = acc10[v];
    xg[r1 * GATES + colBase + 16 + arow] = acc11[v];
  }
}

// ---- Phase 2: persistent recurrent scan.
// Grid = 256 WGs: 4 row-groups (32 batch rows) x 64 unit slices (16 units).
// W_hh slice (4 gates x 16 units x 1024 K = 256KB fp32) lives in LDS for all 256 steps.
__global__ void __launch_bounds__(256) lstm_rec_kernel(
    const float* __restrict__ x, const int* __restrict__ reset,
    const float* __restrict__ Whh, const float* __restrict__ b_ih,
    const float* __restrict__ b_hh, const float* __restrict__ c0,
    const float* __restrict__ xg, float* __restrict__ hping,
    unsigned* counters, float* __restrict__ out) {
  extern __shared__ float smem[];
  float* whh  = smem;                      // 64 x WHH_STRIDE
  float* gbuf = whh + 64 * WHH_STRIDE;     // 32 rows x (4 gates x 16) cols, padded
  float* cbuf = gbuf + 32 * GBUF_STRIDE;   // 32 x 16 cell state
  float* mbuf = cbuf + 32 * 16;            // 32 row masks

  const int tid  = threadIdx.x;
  const int lane = tid & 31;
  const int wave = tid >> 5;   // 0..7
  const int g    = wave & 3;   // gate (i,f,g,o)
  const int rt   = wave >> 2;  // 0..1 row tile
  const int ug = blockIdx.x & 63;
  const int rg = blockIdx.x >> 6;
  const int u0 = ug * 16;
  const int b0 = rg * 32;

  // Stage W_hh slice into LDS (float4 copies; 4112B row stride keeps 16B alignment)
  for (int i = tid; i < 64 * 256; i += 256) {
    int sg = i >> 8;                       // slice row 0..63
    int c4 = i & 255;                      // float4 index along K
    int grow = ((sg >> 4) << 10) + u0 + (sg & 15);  // gate*1024 + u0 + n
    float4 w = *(const float4*)(Whh + (size_t)grow * UNITS + (size_t)c4 * 4);
    *(float4*)(whh + (size_t)sg * WHH_STRIDE + c4 * 4) = w;
  }
  // Cell state
  for (int e = tid; e < 512; e += 256) {
    int r = e >> 4, n = e & 15;
    cbuf[e] = c0[(size_t)(b0 + r) * UNITS + u0 + n];
  }
  // Bias sums (fixed unit column per thread)
  const int n_u = tid & 15;
  float bs[4];
#pragma unroll
  for (int gg = 0; gg < 4; ++gg) {
    int col = gg * UNITS + u0 + n_u;
    bs[gg] = b_ih[col] + b_hh[col];
  }
  __syncthreads();

  const int arow = lane & 15;
  const int khi  = (lane >> 4) << 1;
  const int moff = (lane >> 4) << 3;
  const float* wlane = whh + (size_t)(g * 16 + arow) * WHH_STRIDE + khi;

  for (int t = 0; t < SEQ; ++t) {
    const float* hcur = hping + (size_t)(t & 1) * (BATCH * UNITS);
    float*       hnxt = hping + (size_t)((t + 1) & 1) * (BATCH * UNITS);

    if (tid < 32) mbuf[tid] = 1.0f - (float)reset[t * BATCH + b0 + tid];

    // gate-tile = h(16x1024) @ Whh_slice^T(1024x16); mask folded in later (diag(m)h W = diag(m)(hW))
    const float* ap = hcur + (size_t)(b0 + rt * 16 + arow) * UNITS + khi;
    v8f acc = {};
#pragma unroll 4
    for (int kb = 0; kb < UNITS; kb += 4) {
      v2f a = *(const v2f*)(ap + kb);
      v2f b = *(const v2f*)(wlane + kb);
      acc = wmma4(a, b, acc);
    }
#pragma unroll
    for (int v = 0; v < 8; ++v) {
      int row = rt * 16 + v + moff;
      gbuf[row * GBUF_STRIDE + g * 16 + arow] = acc[v];
    }
    __syncthreads();  // gbuf + mbuf ready

    // Elementwise cell update (32 rows x 16 units -> 2 elems/thread)
    for (int e = tid; e < 512; e += 256) {
      int r = e >> 4, n = e & 15;
      float m = mbuf[r];
      const float* xr = xg + ((size_t)t * BATCH + b0 + r) * GATES + u0 + n;
      float gi = xr[0]         + m * gbuf[r * GBUF_STRIDE + 0  + n] + bs[0];
      float gf = xr[UNITS]     + m * gbuf[r * GBUF_STRIDE + 16 + n] + bs[1];
      float gc = xr[2 * UNITS] + m * gbuf[r * GBUF_STRIDE + 32 + n] + bs[2];
      float go = xr[3 * UNITS] + m * gbuf[r * GBUF_STRIDE + 48 + n] + bs[3];
      float cold = m * cbuf[e];
      float cn = sigm(gf) * cold + sigm(gi) * tanh_fast(gc);
      float hn = sigm(go) * tanh_fast(cn);
      cbuf[e] = cn;
      hnxt[(size_t)(b0 + r) * UNITS + u0 + n] = hn;
      size_t oidx = ((size_t)t * BATCH + b0 + r) * UNITS + u0 + n;
      out[oidx] = hn + x[oidx];  // skip connection (DIM == UNITS)
      if (t == SEQ - 1) {
        out[(size_t)SEQ * BATCH * UNITS + (size_t)(b0 + r) * UNITS + u0 + n] = hn;                         // hT
        out[(size_t)SEQ * BATCH * UNITS + (size_t)BATCH * UNITS + (size_t)(b0 + r) * UNITS + u0 + n] = cn; // cT
      }
    }

    // Device-wide barrier: all WGs finished step t (h[next] fully written, h[cur] fully read)
    __threadfence();
    __syncthreads();
    if (tid == 0) {
      __hip_atomic_fetch_add(&counters[t], 1u, __ATOMIC_ACQ_REL, __HIP_MEMORY_SCOPE_AGENT);
      while (__hip_atomic_load(&counters[t], __ATOMIC_ACQUIRE, __HIP_MEMORY_SCOPE_AGENT) <
             (unsigned)gridDim.x) {
        __builtin_amdgcn_s_sleep(2);
      }
    }
    __syncthreads();
  }
}

extern "C" void kernel_launch(void* const* d_in, const int* in_sizes, int n_in,
                              void* d_out, int out_size, void* d_ws, size_t ws_size,
                              hipStream_t stream) {
  (void)in_sizes; (void)n_in; (void)out_size; (void)ws_size;
  const float* x   = (const float*)d_in[0];
  const int*   rst = (const int*)d_in[1];
  const float* Wih = (const float*)d_in[2];
  const float* Whh = (const float*)d_in[3];
  const float* bih = (const float*)d_in[4];
  const float* bhh = (const float*)d_in[5];
  const float* h0  = (const float*)d_in[6];
  const float* c0  = (const float*)d_in[7];
  float* out = (float*)d_out;

  // workspace layout: [xg: SB*GATES f32 (512 MiB)][h ping-pong: 2*B*U f32][SEQ barrier counters]
  float* xg    = (float*)d_ws;
  float* hping = xg + (size_t)SB * GATES;
  unsigned* counters = (unsigned*)(hping + 2 * BATCH * UNITS);

  init_kernel<<<dim3((BATCH * UNITS + 255) / 256), dim3(256), 0, stream>>>(h0, hping, counters);
  xg_gemm_kernel<<<dim3(SB / 128, GATES / 64), dim3(256), 0, stream>>>(x, Wih, xg);

  const size_t ldsBytes = (size_t)LDS_FLOATS * sizeof(float);  // ~268 KB of the 320 KB WGP LDS
  hipFuncSetAttribute((const void*)lstm_rec_kernel,
                      hipFuncAttributeMaxDynamicSharedMemorySize, (int)ldsBytes);
  lstm_rec_kernel<<<dim3(NBLK_REC), dim3(256), ldsBytes, stream>>>(
      x, rst, Whh, bih, bhh, c0, xg, hping, counters, out);
}